// HandIntersectionLoss_33715493273840
// MI455X (gfx1250) — compile-verified
//
#include <hip/hip_runtime.h>
#include <math.h>

// ---------------------------------------------------------------------------
// HandIntersectionLoss for MI455X (gfx1250, wave32).
//
// Strategy:
//  * 1 workgroup (256 thr = 8 waves) per (frame b, direction dir), 128 blocks.
//  * Stage query points and per-triangle precomputed data (ab, ac, n, dots,
//    a·ab, a·ac, a) in LDS (~64 KB) -> all 451x800 pair tests run on-chip.
//  * d1 = ap·ab and d2 = ap·ac are a rank-3 GEMM P(451x3) x ABt(3x800):
//    computed 16x16 tiles at a time with V_WMMA_F32_16X16X4_F32 (K padded
//    to 4 with zeros), then corrected by per-face constants a·ab / a·ac.
//  * Per lane: fixed triangle column, 8 point rows (WMMA C layout). Scalar
//    Ericson region logic + running (min d2, sign) per row; cross-lane
//    __shfl_xor min over the 16 columns finishes the per-point argmin.
//  * Deterministic reduction: lane0+lane16 -> per-wave LDS slot -> thread 0
//    -> ws partial[block]; tiny finalize kernel sums 128 partials -> out[4].
// ---------------------------------------------------------------------------

typedef __attribute__((ext_vector_type(2))) float v2f;
typedef __attribute__((ext_vector_type(8))) float v8f;

#define EPSF 1e-9f

// Problem sizes fixed by the reference.
#define NVBODY 10475
#define NHAND  450
#define NPTS   451      // hand verts + appended bracelet-mean vertex
#define NBRAC  30
#define NFACE  800
#define NPTILE 29       // ceil(451/16)
#define NFTILE 50       // 800/16
#define TDS    17       // triData stride (floats); gcd(17,64)=1 -> no bank conflicts

__device__ __forceinline__ float sdivf(float num, float den) {
    // matches reference: num / where(|den|>EPS, den, 1.0)
    return num / (fabsf(den) > EPSF ? den : 1.0f);
}

__global__ __launch_bounds__(256) void hand_pen_kernel(
    const float* __restrict__ body,
    const int* __restrict__ vindsL, const int* __restrict__ bindsL, const int* __restrict__ facesL,
    const int* __restrict__ vindsR, const int* __restrict__ bindsR, const int* __restrict__ facesR,
    float* __restrict__ partial)
{
    __shared__ float s_pts[NPTS * 3];     // query points
    __shared__ float s_tv [NPTS * 3];     // triangle-hand vertices
    __shared__ float s_td [NFACE * TDS];  // per-face precomputed data
    __shared__ float s_mean[6];           // bracelet means: [0..2] pt-hand, [3..5] tri-hand
    __shared__ float s_wsum[8];           // per-wave partial sums

    const int blk = blockIdx.x;
    const int b   = blk >> 1;
    const int dir = blk & 1;
    const int tid = threadIdx.x;

    // dir==0: points = LEFT hand, triangles = RIGHT hand (pen(vL, tR)); dir==1 swapped.
    const int* pv = (dir == 0) ? vindsL : vindsR;
    const int* pb = (dir == 0) ? bindsL : bindsR;
    const int* tv = (dir == 0) ? vindsR : vindsL;
    const int* tb = (dir == 0) ? bindsR : bindsL;
    const int* tf = (dir == 0) ? facesR : facesL;
    const float* bod = body + (size_t)b * NVBODY * 3;

    // ---- bracelet means (one coordinate per thread, fixed order -> deterministic)
    if (tid < 6) {
        const int* binds = (tid < 3) ? pb : tb;
        const int  c     = tid % 3;
        float s = 0.0f;
        for (int j = 0; j < NBRAC; ++j) s += bod[(size_t)binds[j] * 3 + c];
        s_mean[tid] = s * (1.0f / (float)NBRAC);
    }
    __syncthreads();

    // ---- gather query points and triangle-hand vertices into LDS
    for (int i = tid; i < NPTS; i += 256) {
        float x, y, z;
        if (i < NHAND) { size_t g = (size_t)pv[i] * 3; x = bod[g]; y = bod[g+1]; z = bod[g+2]; }
        else           { x = s_mean[0]; y = s_mean[1]; z = s_mean[2]; }
        s_pts[i*3+0] = x; s_pts[i*3+1] = y; s_pts[i*3+2] = z;
    }
    for (int i = tid; i < NPTS; i += 256) {
        float x, y, z;
        if (i < NHAND) { size_t g = (size_t)tv[i] * 3; x = bod[g]; y = bod[g+1]; z = bod[g+2]; }
        else           { x = s_mean[3]; y = s_mean[4]; z = s_mean[5]; }
        s_tv[i*3+0] = x; s_tv[i*3+1] = y; s_tv[i*3+2] = z;
    }
    __syncthreads();

    // ---- per-face precompute: ab, ac, unit n, abab/abac/acac, a·ab, a·ac, a
    for (int f = tid; f < NFACE; f += 256) {
        int i0 = tf[f*3+0], i1 = tf[f*3+1], i2 = tf[f*3+2];
        float ax = s_tv[i0*3], ay = s_tv[i0*3+1], az = s_tv[i0*3+2];
        float bx = s_tv[i1*3], by = s_tv[i1*3+1], bz = s_tv[i1*3+2];
        float cx = s_tv[i2*3], cy = s_tv[i2*3+1], cz = s_tv[i2*3+2];
        float abx = bx-ax, aby = by-ay, abz = bz-az;
        float acx = cx-ax, acy = cy-ay, acz = cz-az;
        float nx = aby*acz - abz*acy;
        float ny = abz*acx - abx*acz;
        float nz = abx*acy - aby*acx;
        float nl  = sqrtf(nx*nx + ny*ny + nz*nz);
        float inv = 1.0f / fmaxf(nl, EPSF);
        nx *= inv; ny *= inv; nz *= inv;
        float* td = &s_td[f * TDS];
        td[0]=abx; td[1]=aby; td[2]=abz;
        td[3]=acx; td[4]=acy; td[5]=acz;
        td[6]=nx;  td[7]=ny;  td[8]=nz;
        td[9]  = abx*abx + aby*aby + abz*abz;     // abab
        td[10] = abx*acx + aby*acy + abz*acz;     // abac
        td[11] = acx*acx + acy*acy + acz*acz;     // acac
        td[12] = ax*abx + ay*aby + az*abz;        // a·ab
        td[13] = ax*acx + ay*acy + az*acz;        // a·ac
        td[14] = ax; td[15] = ay; td[16] = az;
    }
    __syncthreads();

    // ---- main pair loop: wave owns a 16-point tile, sweeps 50 tri-tiles
    const int  wave = tid >> 5;
    const int  lane = tid & 31;
    const int  col  = lane & 15;   // triangle column within tile
    const bool hi   = lane >= 16;  // C layout: hi lanes hold rows 8..15

    float mysum = 0.0f;

    for (int ptile = wave; ptile < NPTILE; ptile += 8) {
        // A matrix (16x4 f32): lanes 0-15 -> (x,y) K=0,1 ; lanes 16-31 -> (z,0) K=2,3
        int Mi = ptile*16 + col; if (Mi > NPTS-1) Mi = NPTS-1;  // clamp padding
        v2f A;
        if (!hi) { A.x = s_pts[Mi*3+0]; A.y = s_pts[Mi*3+1]; }
        else     { A.x = s_pts[Mi*3+2]; A.y = 0.0f; }

        // preload this lane's 8 point rows
        float px[8], py[8], pz[8];
        #pragma unroll
        for (int m = 0; m < 8; ++m) {
            int pidx = ptile*16 + m + (hi ? 8 : 0);
            int pc   = (pidx > NPTS-1) ? (NPTS-1) : pidx;
            px[m] = s_pts[pc*3+0]; py[m] = s_pts[pc*3+1]; pz[m] = s_pts[pc*3+2];
        }

        float bestd2[8], bests[8];
        #pragma unroll
        for (int m = 0; m < 8; ++m) { bestd2[m] = 3.0e38f; bests[m] = 1.0f; }

        for (int ftile = 0; ftile < NFTILE; ++ftile) {
            const float* td = &s_td[(ftile*16 + col) * TDS];
            float abx=td[0],  aby=td[1],  abz=td[2];
            float acx=td[3],  acy=td[4],  acz=td[5];
            float nx =td[6],  ny =td[7],  nz =td[8];
            float abab=td[9], abac=td[10], acac=td[11];
            float aab=td[12], aac=td[13];
            float ax =td[14], ay =td[15], az =td[16];

            // B matrices (4x16 f32): lanes 0-15 -> K=0,1 ; lanes 16-31 -> K=2,3(=0 pad)
            v2f B1, B2;
            if (!hi) { B1.x=abx; B1.y=aby; B2.x=acx; B2.y=acy; }
            else     { B1.x=abz; B1.y=0.f; B2.x=acz; B2.y=0.f; }

            v8f zero = {};
            // D[m][n] = p_m · ab_n  (and · ac_n) via rank-3 WMMA (K padded to 4)
            v8f D1 = __builtin_amdgcn_wmma_f32_16x16x4_f32(
                         false, A, false, B1, (short)0, zero, false, false);
            v8f D2 = __builtin_amdgcn_wmma_f32_16x16x4_f32(
                         false, A, false, B2, (short)0, zero, false, false);

            #pragma unroll
            for (int m = 0; m < 8; ++m) {
                float d1 = D1[m] - aab;          // ap·ab
                float d2 = D2[m] - aac;          // ap·ac
                float d3 = d1 - abab;
                float d4 = d2 - abac;
                float d5 = d1 - abac;
                float d6 = d2 - acac;
                float vc = d1*d4 - d3*d2;
                float vb = d5*d2 - d1*d6;
                float va = d3*d6 - d5*d4;

                float t4 = sdivf(d1, d1 - d3);
                float t5 = sdivf(d2, d2 - d6);
                float t6 = sdivf(d4 - d3, (d4 - d3) + (d5 - d6));
                float den = sdivf(1.0f, va + vb + vc);
                float v = vb * den;
                float w = vc * den;
                // region overrides, reference where-order (r1 highest priority)
                if (va <= 0.f && (d4 - d3) >= 0.f && (d5 - d6) >= 0.f) { v = 1.f - t6; w = t6; } // r6
                if (vb <= 0.f && d2 >= 0.f && d6 <= 0.f)               { v = 0.f;      w = t5; } // r5
                if (vc <= 0.f && d1 >= 0.f && d3 <= 0.f)               { v = t4;       w = 0.f;} // r4
                if (d6 >= 0.f && d5 <= d6)                             { v = 0.f;      w = 1.f;} // r3
                if (d3 >= 0.f && d4 <= d3)                             { v = 1.f;      w = 0.f;} // r2
                if (d1 <= 0.f && d2 <= 0.f)                            { v = 0.f;      w = 0.f;} // r1

                // e = p - closest = ap - v*ab - w*ac
                float ex = (px[m]-ax) - v*abx - w*acx;
                float ey = (py[m]-ay) - v*aby - w*acy;
                float ez = (pz[m]-az) - v*abz - w*acz;
                float dist2 = ex*ex + ey*ey + ez*ez;
                float sgn   = ex*nx + ey*ny + ez*nz;
                if (dist2 < bestd2[m]) { bestd2[m] = dist2; bests[m] = sgn; }
            }
        }

        // cross-lane min (with sign payload) over the 16 triangle columns
        #pragma unroll
        for (int m = 0; m < 8; ++m) {
            float d2v = bestd2[m], sv = bests[m];
            #pragma unroll
            for (int off = 1; off < 16; off <<= 1) {
                float od = __shfl_xor(d2v, off, 32);
                float os = __shfl_xor(sv,  off, 32);
                if (od < d2v) { d2v = od; sv = os; }
            }
            int pidx = ptile*16 + m + (hi ? 8 : 0);
            if (col == 0 && pidx < NPTS && sv < 0.0f) {
                mysum += sqrtf(fmaxf(d2v, EPSF));
            }
        }
    }

    // ---- deterministic block reduction
    float other = __shfl(mysum, 16, 32);           // lane 16's sum
    if (lane == 0) s_wsum[wave] = mysum + other;
    __syncthreads();
    if (tid == 0) {
        float s = 0.0f;
        for (int wv = 0; wv < 8; ++wv) s += s_wsum[wv];
        partial[blk] = s;
    }
}

__global__ void hand_pen_finalize(const float* __restrict__ partial,
                                  float* __restrict__ out, int nout, int seq)
{
    int i = threadIdx.x;
    if (i < nout) {
        float s = 0.0f;
        for (int bb = i * seq; bb < (i + 1) * seq; ++bb)
            s += partial[2*bb] + partial[2*bb + 1];
        out[i] = s / (float)seq;
    }
}

extern "C" void kernel_launch(void* const* d_in, const int* in_sizes, int n_in,
                              void* d_out, int out_size, void* d_ws, size_t ws_size,
                              hipStream_t stream) {
    const float* body = (const float*)d_in[0];
    const int* vL = (const int*)d_in[1];
    const int* bL = (const int*)d_in[2];
    const int* fL = (const int*)d_in[3];
    const int* vR = (const int*)d_in[4];
    const int* bR = (const int*)d_in[5];
    const int* fR = (const int*)d_in[6];
    float* out     = (float*)d_out;
    float* partial = (float*)d_ws;   // 2*nbatch floats

    const int nbatch = in_sizes[0] / (NVBODY * 3);      // bs*seq = 64
    const int seq    = nbatch / (out_size > 0 ? out_size : 1);

    hand_pen_kernel<<<nbatch * 2, 256, 0, stream>>>(body, vL, bL, fL, vR, bR, fR, partial);
    hand_pen_finalize<<<1, 64, 0, stream>>>(partial, out, out_size, seq);
}